// UHGAttentionLayer_21328807592548
// MI455X (gfx1250) — compile-verified
//
#include <hip/hip_runtime.h>

#define IN_F 128
#define OUT_F 64
#define NHEAD 4
#define HDIM 16

typedef __attribute__((ext_vector_type(2))) float v2f;
typedef __attribute__((ext_vector_type(8))) float v8f;

// out[M x 64] = in[M x kf] @ W[64 x kf]^T + bias, fp32 WMMA 16x16x4.
// One wave (32 lanes) per 16-row tile; 4 column tiles of 16 (== one head each).
// A layout (16x4 f32): lane l holds M = l&15, VGPR j holds K = j + 2*(l>>4).
// B layout (4x16 f32): lane l holds N = l&15, VGPR j holds K = j + 2*(l>>4).
// C/D layout (16x16 f32): lane l holds N = l&15, VGPR j holds M = j + 8*(l>>4).
__global__ void gemm_bias_wmma(const float* __restrict__ in,
                               const float* __restrict__ W,
                               const float* __restrict__ bias,
                               float* __restrict__ out,
                               int n_rows, int kf)
{
    const int lane    = threadIdx.x;                        // 0..31
    const int rowTile = blockIdx.x * blockDim.y + threadIdx.y;
    const int r0      = rowTile * 16;
    if (r0 >= n_rows) return;                               // wave-uniform guard (EXEC stays full)

    const int ln   = lane & 15;
    const int half = lane >> 4;
    const int kb   = half * 2;

    const float* arow = in + (size_t)(r0 + ln) * kf + kb;

    for (int nt = 0; nt < 4; ++nt) {
        const int n0 = nt * 16;
        const float* brow = W + (size_t)(n0 + ln) * kf + kb;
        v8f acc = {};
        for (int k0 = 0; k0 < kf; k0 += 4) {
            v2f a = *(const v2f*)(arow + k0);
            v2f b = *(const v2f*)(brow + k0);
            acc = __builtin_amdgcn_wmma_f32_16x16x4_f32(
                false, a, false, b, (short)0, acc, false, false);
        }
        const float bn = bias[n0 + ln];
        float* obase = out + (size_t)(r0 + half * 8) * OUT_F + n0 + ln;
#pragma unroll
        for (int j = 0; j < 8; ++j)
            obase[(size_t)j * OUT_F] = acc[j] + bn;
    }
}

__device__ __forceinline__ float mink_self(const float4* a)
{
    float4 a0 = a[0], a1 = a[1], a2 = a[2], a3 = a[3];
    return a0.x*a0.x + a0.y*a0.y + a0.z*a0.z + a0.w*a0.w
         + a1.x*a1.x + a1.y*a1.y + a1.z*a1.z + a1.w*a1.w
         + a2.x*a2.x + a2.y*a2.y + a2.z*a2.z + a2.w*a2.w
         + a3.x*a3.x + a3.y*a3.y + a3.z*a3.z - a3.w*a3.w;
}

// Per-(node,head) Minkowski self inner products for q and k.
__global__ void node_norms(const float* __restrict__ q, const float* __restrict__ k,
                           float* __restrict__ qq, float* __restrict__ kk, int NH)
{
    int t = blockIdx.x * blockDim.x + threadIdx.x;
    if (t >= NH) return;
    qq[t] = mink_self((const float4*)(q + (size_t)t * HDIM));
    kk[t] = mink_self((const float4*)(k + (size_t)t * HDIM));
}

// Per-(edge,head) UHG score + segment-max via unsigned atomicMin (scores < 0).
__global__ void edge_scores(const float* __restrict__ q, const float* __restrict__ k,
                            const float* __restrict__ qq, const float* __restrict__ kk,
                            const int* __restrict__ src, const int* __restrict__ dst,
                            float* __restrict__ scores, unsigned* __restrict__ smax, int EH)
{
    int t = blockIdx.x * blockDim.x + threadIdx.x;
    if (t >= EH) return;
    int e = t >> 2, h = t & 3;
    int s = src[e], d = dst[e];

    const float4* qa = (const float4*)(q + (size_t)s * OUT_F + h * HDIM);
    const float4* ka = (const float4*)(k + (size_t)d * OUT_F + h * HDIM);
    float4 q0 = qa[0], q1 = qa[1], q2 = qa[2], q3 = qa[3];
    float4 k0 = ka[0], k1 = ka[1], k2 = ka[2], k3 = ka[3];

    float dot = q0.x*k0.x + q0.y*k0.y + q0.z*k0.z + q0.w*k0.w
              + q1.x*k1.x + q1.y*k1.y + q1.z*k1.z + q1.w*k1.w
              + q2.x*k2.x + q2.y*k2.y + q2.z*k2.z + q2.w*k2.w
              + q3.x*k3.x + q3.y*k3.y + q3.z*k3.z - q3.w*k3.w;

    float num   = dot * dot;
    float den   = qq[s * NHEAD + h] * kk[d * NHEAD + h];
    float ratio = num / fmaxf(fabsf(den), 1e-9f);
    float x     = fmaxf(sqrtf(ratio + 1e-9f), 1.0f + 1e-6f);
    float dist  = logf(x + sqrtf(fmaf(x, x, -1.0f)));   // arccosh(x)
    float sc    = -dist;

    scores[t] = sc;
    // sc < 0 always: max over negative floats == min over unsigned bit patterns
    atomicMin(&smax[d * NHEAD + h], __float_as_uint(sc));
}

__global__ void edge_exp(float* __restrict__ scores, const unsigned* __restrict__ smax,
                         float* __restrict__ denom, const int* __restrict__ dst, int EH)
{
    int t = blockIdx.x * blockDim.x + threadIdx.x;
    if (t >= EH) return;
    int e = t >> 2, h = t & 3;
    int d = dst[e];
    float m  = __uint_as_float(smax[d * NHEAD + h]);
    float ex = expf(scores[t] - m);
    scores[t] = ex;                         // in-place: scores buffer becomes exp buffer
    atomicAdd(&denom[d * NHEAD + h], ex);
}

__global__ void edge_agg(const float* __restrict__ ex, const float* __restrict__ denom,
                         const float* __restrict__ v,
                         const int* __restrict__ src, const int* __restrict__ dst,
                         float* __restrict__ agg, int EH)
{
    int t = blockIdx.x * blockDim.x + threadIdx.x;
    if (t >= EH) return;
    int e = t >> 2, h = t & 3;
    int s = src[e], d = dst[e];
    float w = ex[t] / fmaxf(denom[d * NHEAD + h], 1e-9f);
    const float* vs = v + (size_t)s * OUT_F + h * HDIM;
    float* ag = agg + (size_t)d * OUT_F + h * HDIM;
#pragma unroll
    for (int i = 0; i < HDIM; ++i)
        atomicAdd(&ag[i], w * vs[i]);
}

// Per-(node,head) L2 normalization: attended = agg / (||agg|| + eps)
__global__ void normalize_agg(const float* __restrict__ agg, float* __restrict__ att, int NH)
{
    int t = blockIdx.x * blockDim.x + threadIdx.x;
    if (t >= NH) return;
    const float4* a = (const float4*)(agg + (size_t)t * HDIM);
    float4 a0 = a[0], a1 = a[1], a2 = a[2], a3 = a[3];
    float ss = a0.x*a0.x + a0.y*a0.y + a0.z*a0.z + a0.w*a0.w
             + a1.x*a1.x + a1.y*a1.y + a1.z*a1.z + a1.w*a1.w
             + a2.x*a2.x + a2.y*a2.y + a2.z*a2.z + a2.w*a2.w
             + a3.x*a3.x + a3.y*a3.y + a3.z*a3.z + a3.w*a3.w;
    float inv = 1.0f / (sqrtf(ss) + 1e-9f);
    float4* o = (float4*)(att + (size_t)t * HDIM);
    o[0] = make_float4(a0.x*inv, a0.y*inv, a0.z*inv, a0.w*inv);
    o[1] = make_float4(a1.x*inv, a1.y*inv, a1.z*inv, a1.w*inv);
    o[2] = make_float4(a2.x*inv, a2.y*inv, a2.z*inv, a2.w*inv);
    o[3] = make_float4(a3.x*inv, a3.y*inv, a3.z*inv, a3.w*inv);
}

extern "C" void kernel_launch(void* const* d_in, const int* in_sizes, int n_in,
                              void* d_out, int out_size, void* d_ws, size_t ws_size,
                              hipStream_t stream)
{
    const float* x  = (const float*)d_in[0];
    const int*   ei = (const int*)d_in[1];     // [2, E]: src row then dst row
    const float* Wq = (const float*)d_in[2];
    const float* bq = (const float*)d_in[3];
    const float* Wk = (const float*)d_in[4];
    const float* bk = (const float*)d_in[5];
    const float* Wv = (const float*)d_in[6];
    const float* bv = (const float*)d_in[7];
    const float* Wo = (const float*)d_in[8];
    const float* bo = (const float*)d_in[9];
    float* out = (float*)d_out;

    const int N  = in_sizes[0] / IN_F;         // 50000 (multiple of 16)
    const int E  = in_sizes[1] / 2;            // 800000
    const int NH = N * NHEAD;
    const int EH = E * NHEAD;
    const int* src = ei;
    const int* dst = ei + E;

    // workspace carve-up (all offsets 16B-aligned)
    float*    q      = (float*)d_ws;                    // N*64 (reused as "attended")
    float*    kbuf   = q    + (size_t)N * OUT_F;        // N*64
    float*    vbuf   = kbuf + (size_t)N * OUT_F;        // N*64
    float*    qq     = vbuf + (size_t)N * OUT_F;        // N*4
    float*    kk     = qq   + NH;                       // N*4
    unsigned* smax   = (unsigned*)(kk + NH);            // N*4
    float*    denom  = (float*)(smax + NH);             // N*4
    float*    agg    = denom + NH;                      // N*64
    float*    scores = agg + (size_t)N * OUT_F;         // E*4

    hipMemsetAsync(smax,  0xFF, (size_t)NH * sizeof(unsigned), stream); // unsigned-min identity
    hipMemsetAsync(denom, 0,    (size_t)NH * sizeof(float), stream);
    hipMemsetAsync(agg,   0,    (size_t)N * OUT_F * sizeof(float), stream);

    dim3 gblk(32, 4);                                   // 4 waves, one 16-row tile each
    int  rowBlocks = (N / 16 + 3) / 4;

    gemm_bias_wmma<<<rowBlocks, gblk, 0, stream>>>(x, Wq, bq, q,    N, IN_F);
    gemm_bias_wmma<<<rowBlocks, gblk, 0, stream>>>(x, Wk, bk, kbuf, N, IN_F);
    gemm_bias_wmma<<<rowBlocks, gblk, 0, stream>>>(x, Wv, bv, vbuf, N, IN_F);

    node_norms   <<<(NH + 255) / 256, 256, 0, stream>>>(q, kbuf, qq, kk, NH);
    edge_scores  <<<(EH + 255) / 256, 256, 0, stream>>>(q, kbuf, qq, kk, src, dst, scores, smax, EH);
    edge_exp     <<<(EH + 255) / 256, 256, 0, stream>>>(scores, smax, denom, dst, EH);
    edge_agg     <<<(EH + 255) / 256, 256, 0, stream>>>(scores, denom, vbuf, src, dst, agg, EH);
    normalize_agg<<<(NH + 255) / 256, 256, 0, stream>>>(agg, q, NH);   // attended -> q buffer

    gemm_bias_wmma<<<rowBlocks, gblk, 0, stream>>>(q, Wo, bo, out, N, OUT_F);
}